// GoM_19069654794830
// MI455X (gfx1250) — compile-verified
//
#include <hip/hip_runtime.h>

// 1-NN drift lookup via V_WMMA_F32_16X16X4_F32 on gfx1250 (wave32).
//
// Score: d2[m][n] = ||g_n||^2 - 2*(y_m . g_n)   ==  A(16x4) x B(4x16)
//   A row m: [-2*y_m.x, -2*y_m.y, 1, 0]     (particles, held in regs all kernel)
//   B col n: [g_n.x, ||g_n||^2, g_n.y,    0]  (rebuilt per 16-column tile)
//
// ISA A-fragment layout (32-bit A 16x4): v0 = K0 (lanes 0-15) | K2 (lanes 16-31)
//                                        v1 = K1 (lanes 0-15) | K3 (lanes 16-31)
// B-fragment (4x16) row-striped the same way.
// C/D: lane holds column N=lane%16; VGPR r holds row M = r + 8*(lane>=16).

typedef float v2f __attribute__((ext_vector_type(2)));
typedef float v8f __attribute__((ext_vector_type(8)));

__global__ __launch_bounds__(256)
void nn_drift_wmma_kernel(const float* __restrict__ y,
                          const float* __restrict__ gp,
                          const float* __restrict__ ug,
                          const float* __restrict__ vg,
                          const float* __restrict__ log_sigma,
                          float* __restrict__ out,
                          int n_particles, int n_grid)
{
    const int lane       = (int)(threadIdx.x & 31u);
    const int wave_id    = (int)(blockIdx.x * (blockDim.x >> 5) + (threadIdx.x >> 5));
    const int rowbase    = wave_id * 16;            // 16 particles per wave
    if (rowbase >= n_particles) return;             // wave-uniform: EXEC stays all-1s

    const int  lane16 = lane & 15;
    const bool hi     = lane >= 16;

    // ---- A fragment: one particle row per lane group, resident all kernel ----
    const float2 yv = ((const float2*)y)[rowbase + lane16];
    v2f a;
    a.x = hi ? 1.0f : (-2.0f * yv.x);   // K0 | K2
    a.y = hi ? 0.0f : (-2.0f * yv.y);   // K1 | K3

    float minv[8];
    int   mini[8];
    #pragma unroll
    for (int r = 0; r < 8; ++r) { minv[r] = __builtin_inff(); mini[r] = 0; }

    // ---- sweep all grid columns, 16 per WMMA ----
    const int ntiles = n_grid >> 4;
    for (int ct = 0; ct < ntiles; ++ct) {
        const int col   = (ct << 4) + lane16;       // this lane's column
        const float2 g  = ((const float2*)gp)[col]; // one b64 load per lane
        v2f b;
        b.x = hi ? g.y  : g.x;                           // K0 | K2  (gx | gy)
        b.y = hi ? 0.0f : fmaf(g.x, g.x, g.y * g.y);     // K1 | K3  (||g||^2 | 0)

        v8f c = {0.f, 0.f, 0.f, 0.f, 0.f, 0.f, 0.f, 0.f};
        c = __builtin_amdgcn_wmma_f32_16x16x4_f32(
                /*neg_a=*/false, a, /*neg_b=*/false, b,
                /*c_mod=*/(short)0, c, /*reuse_a=*/false, /*reuse_b=*/false);

        // per-lane running argmin over its column stream (strict < keeps first idx)
        #pragma unroll
        for (int r = 0; r < 8; ++r) {
            const float d2   = c[r];
            const bool  take = d2 < minv[r];
            minv[r] = take ? d2  : minv[r];
            mini[r] = take ? col : mini[r];
        }
    }

    // ---- reduce across the 16 lanes of each half (rows 0-7 | rows 8-15) ----
    // XOR masks 1,2,4,8 permute strictly within each 16-lane half of a wave32.
    // Branchless tie-break (lower column index wins, matching jnp.argmin).
    #pragma unroll
    for (int r = 0; r < 8; ++r) {
        float mv = minv[r];
        int   mi = mini[r];
        #pragma unroll
        for (int mask = 1; mask <= 8; mask <<= 1) {
            const float ov   = __shfl_xor(mv, mask, 32);
            const int   oi   = __shfl_xor(mi, mask, 32);
            const bool  take = (ov < mv) | ((ov == mv) & (oi < mi));
            mv = take ? ov : mv;
            mi = take ? oi : mi;
        }
        minv[r] = mv;
        mini[r] = mi;
    }

    // ---- epilogue: lane 0 -> rows 0..7, lane 16 -> rows 8..15 ----
    const float sigma = __expf(log_sigma[0]);
    if (lane16 == 0) {
        const int mbase = rowbase + (hi ? 8 : 0);
        #pragma unroll
        for (int r = 0; r < 8; ++r) {
            const int row = mbase + r;
            const int idx = mini[r];
            ((float2*)out)[row]                     = make_float2(ug[idx], vg[idx]);
            ((float2*)(out + 2 * n_particles))[row] = make_float2(sigma, sigma);
        }
    }
}

extern "C" void kernel_launch(void* const* d_in, const int* in_sizes, int n_in,
                              void* d_out, int out_size, void* d_ws, size_t ws_size,
                              hipStream_t stream) {
    const float* y  = (const float*)d_in[0];   // [N,2]
    // d_in[1] = t (unused by reference math)
    const float* gp = (const float*)d_in[2];   // [G,2]
    const float* ug = (const float*)d_in[3];   // [G]
    const float* vg = (const float*)d_in[4];   // [G]
    const float* ls = (const float*)d_in[5];   // [1]
    float* out = (float*)d_out;                // drift [N,2] ++ diffusion [N,2]

    const int n_particles = in_sizes[0] / 2;
    const int n_grid      = in_sizes[3];

    const int waves  = (n_particles + 15) / 16;   // one wave per 16-particle tile
    const int blocks = (waves + 7) / 8;           // 8 waves (256 threads) per block
    nn_drift_wmma_kernel<<<blocks, 256, 0, stream>>>(y, gp, ug, vg, ls, out,
                                                     n_particles, n_grid);
}